// OPTAlphaModel_73229192397413
// MI455X (gfx1250) — compile-verified
//
#include <hip/hip_runtime.h>
#include <math.h>

// ---------------------------------------------------------------------------
// OPT alpha-model hyper-gradient loop for MI455X (gfx1250).
//
// Per step t (500 steps, hard sequential chain through theta):
//   KA : z = Xn @ theta (WMMA f32 16x16x4), coef = alpha_t*(sigmoid(z)-y)
//        PLUS dev loss L(theta) over dev_xn (same WMMA matvec + reduction)
//   K2 : grad = Xn^T @ coef ; theta -= ETA*grad   (deterministic LDS reduce)
// Working set (xn 32MB + dev_xn 64MB) is L2-resident (192MB L2), so the loop
// streams ~128MB/step from L2.  fp32 everywhere (WMMA_F32_16X16X4_F32).
// No atomics -> bit-deterministic across graph replays.
// ---------------------------------------------------------------------------

typedef __attribute__((ext_vector_type(2))) float v2f;
typedef __attribute__((ext_vector_type(8))) float v8f;

#define DIMK    2048
#define NTRAIN  4096
#define MDEV    8192
#define TSTEPS  500
#define ETA     0.1f

// 16-row fp32 matvec block: returns C where every column of the 16x16 C tile
// equals z (because B's 16 columns are all the same theta chunk).
// Per the CDNA5 ISA A-layout for 16x4 f32: lane L holds A[L&15][2h + {0,1}],
// h = L>>4.  B mirrors that K mapping: (half h, vgpr v) -> k = 2h + v.
__device__ __forceinline__ v8f rowblock_matvec(const float* __restrict__ row_ptr,
                                               const float* __restrict__ theta,
                                               int h) {
  v8f c0 = {}, c1 = {}, c2 = {}, c3 = {};
  const int ko = 2 * h;
#pragma unroll 2
  for (int k = 0; k < DIMK; k += 16) {
    v2f a0 = *(const v2f*)(row_ptr + k + 0  + ko);
    v2f a1 = *(const v2f*)(row_ptr + k + 4  + ko);
    v2f a2 = *(const v2f*)(row_ptr + k + 8  + ko);
    v2f a3 = *(const v2f*)(row_ptr + k + 12 + ko);
    v2f b0 = *(const v2f*)(theta   + k + 0  + ko);
    v2f b1 = *(const v2f*)(theta   + k + 4  + ko);
    v2f b2 = *(const v2f*)(theta   + k + 8  + ko);
    v2f b3 = *(const v2f*)(theta   + k + 12 + ko);
    c0 = __builtin_amdgcn_wmma_f32_16x16x4_f32(false, a0, false, b0, (short)0, c0, false, false);
    c1 = __builtin_amdgcn_wmma_f32_16x16x4_f32(false, a1, false, b1, (short)0, c1, false, false);
    c2 = __builtin_amdgcn_wmma_f32_16x16x4_f32(false, a2, false, b2, (short)0, c2, false, false);
    c3 = __builtin_amdgcn_wmma_f32_16x16x4_f32(false, a3, false, b3, (short)0, c3, false, false);
  }
  return c0 + c1 + c2 + c3;
}

// Fused kernel: blocks [0,32) compute coef for step `slot` (skipped when
// slot == TSTEPS); blocks [32,96) compute dev-loss partials for theta at
// `slot` into loss_part[slot*64 + devBlock].
__global__ void __launch_bounds__(256)
ka_kernel(const float* __restrict__ xn, const float* __restrict__ yn,
          const float* __restrict__ dev_xn, const float* __restrict__ dev_yn,
          const float* __restrict__ alphas, const float* __restrict__ theta,
          float* __restrict__ coef, float* __restrict__ loss_part, int slot) {
  const int tid  = threadIdx.x;
  const int wave = tid >> 5;
  const int lane = tid & 31;
  const int h    = lane >> 4;
  const int b    = blockIdx.x;

  if (b < 32) {
    if (slot >= TSTEPS) return;                 // no alpha row for slot==T
    const int r0  = b * 128 + wave * 16;
    const int row = r0 + (lane & 15);
    v8f c = rowblock_matvec(xn + (size_t)row * DIMK, theta, h);
    if ((lane & 15) == 0) {
      const float* arow = alphas + (size_t)slot * NTRAIN;
#pragma unroll
      for (int r = 0; r < 8; ++r) {
        const int rw = r0 + 8 * h + r;
        const float z = c[r];
        const float s = 1.0f / (1.0f + expf(-z));
        coef[rw] = arow[rw] * (s - yn[rw]);
      }
    }
  } else {
    __shared__ float red[256];
    const int db  = b - 32;
    const int r0  = db * 128 + wave * 16;
    const int row = r0 + (lane & 15);
    v8f c = rowblock_matvec(dev_xn + (size_t)row * DIMK, theta, h);
    float partial = 0.0f;
    if ((lane & 15) == 0) {
#pragma unroll
      for (int r = 0; r < 8; ++r) {
        const int rw = r0 + 8 * h + r;
        const float z = c[r];
        const float y = dev_yn[rw];
        float l = -y * z + log1pf(expf(z));
        if (isinf(l)) l = (1.0f - y) * z;       // match reference inf guard
        partial += l;
      }
    }
    red[tid] = partial;
    __syncthreads();
    for (int off = 128; off > 0; off >>= 1) {
      if (tid < off) red[tid] += red[tid + off];
      __syncthreads();
    }
    if (tid == 0) loss_part[slot * 64 + db] = red[0];
  }
}

// grad[d] = sum_n coef[n]*xn[n][d];  theta[d] -= ETA*grad[d].
// 32 blocks x 256 threads: block owns 64 columns, 4 threads per column split
// the n-range; deterministic LDS reduction (no atomics).
__global__ void __launch_bounds__(256)
grad_update_kernel(const float* __restrict__ xn, const float* __restrict__ coef,
                   float* __restrict__ theta) {
  __shared__ float red[256];
  const int tid = threadIdx.x;
  const int dL  = tid & 63;
  const int nc  = tid >> 6;
  const int d   = blockIdx.x * 64 + dL;
  const int n0  = nc * (NTRAIN / 4);
  float sum = 0.0f;
  for (int n = n0; n < n0 + NTRAIN / 4; n += 4) {
    sum += coef[n + 0] * xn[(size_t)(n + 0) * DIMK + d];
    sum += coef[n + 1] * xn[(size_t)(n + 1) * DIMK + d];
    sum += coef[n + 2] * xn[(size_t)(n + 2) * DIMK + d];
    sum += coef[n + 3] * xn[(size_t)(n + 3) * DIMK + d];
  }
  red[tid] = sum;
  __syncthreads();
  if (nc == 0) {
    const float tot = red[dL] + red[dL + 64] + red[dL + 128] + red[dL + 192];
    theta[d] -= ETA * tot;
  }
}

__global__ void init_kernel(const float* __restrict__ theta_in,
                            float* __restrict__ theta_ws) {
  const int i = blockIdx.x * blockDim.x + threadIdx.x;
  if (i < DIMK) theta_ws[i] = theta_in[i];
}

// out[0] = mean over t=1..500 of L(theta_t); out[1+t] = L(theta_t), t=0..499.
__global__ void __launch_bounds__(256)
final_kernel(const float* __restrict__ loss_part, float* __restrict__ out) {
  __shared__ float red[256];
  const int tid = threadIdx.x;
  float acc = 0.0f;
  for (int t = tid; t <= TSTEPS; t += 256) {
    float s = 0.0f;
    for (int j = 0; j < 64; ++j) s += loss_part[t * 64 + j];
    const float L = s * (1.0f / (float)MDEV);
    if (t < TSTEPS) out[1 + t] = L;             // full_losses[t] = L(theta_t)
    if (t >= 1) acc += L;                       // i_losses sum
  }
  red[tid] = acc;
  __syncthreads();
  for (int off = 128; off > 0; off >>= 1) {
    if (tid < off) red[tid] += red[tid + off];
    __syncthreads();
  }
  if (tid == 0) out[0] = red[0] * (1.0f / (float)TSTEPS);
}

extern "C" void kernel_launch(void* const* d_in, const int* in_sizes, int n_in,
                              void* d_out, int out_size, void* d_ws, size_t ws_size,
                              hipStream_t stream) {
  const float* theta_in = (const float*)d_in[0];
  const float* alphas   = (const float*)d_in[1];
  const float* xn       = (const float*)d_in[2];
  const float* yn       = (const float*)d_in[3];
  const float* dev_xn   = (const float*)d_in[4];
  const float* dev_yn   = (const float*)d_in[5];
  float* out = (float*)d_out;

  float* theta     = (float*)d_ws;              // 2048
  float* coef      = theta + DIMK;              // 4096
  float* loss_part = coef + NTRAIN;             // 501*64

  init_kernel<<<8, 256, 0, stream>>>(theta_in, theta);
  // loss slot 0 = L(theta_0) and coef for step 0
  ka_kernel<<<96, 256, 0, stream>>>(xn, yn, dev_xn, dev_yn, alphas, theta,
                                    coef, loss_part, 0);
  for (int t = 0; t < TSTEPS; ++t) {
    grad_update_kernel<<<32, 256, 0, stream>>>(xn, coef, theta);
    // loss slot t+1 = L(theta_{t+1}); coef for step t+1 (skipped at t+1==T)
    ka_kernel<<<96, 256, 0, stream>>>(xn, yn, dev_xn, dev_yn, alphas, theta,
                                      coef, loss_part, t + 1);
  }
  final_kernel<<<1, 256, 0, stream>>>(loss_part, out);
}